// TopKSAE_4097398800656
// MI455X (gfx1250) — compile-verified
//
#include <hip/hip_runtime.h>
#include <hip/hip_bf16.h>

// ---------------------------------------------------------------------------
// TopK SAE forward for MI455X (gfx1250, wave32).
//   z     = x @ W_enc^T + b_enc          [8192, 16384]
//   z     = topk_mask(z, k=32) (by |z|)
//   recon = z @ W_dec^T + b_dec          [8192, 1024]
// d_out = [recon | z] flat.
// ---------------------------------------------------------------------------

#define N_TOKENS 8192
#define D_MODEL  1024
#define D_DICT   16384
#define TOPK     32

typedef __attribute__((ext_vector_type(2))) float v2f;
typedef __attribute__((ext_vector_type(8))) float v8f;
typedef __attribute__((ext_vector_type(4))) int v4i;

// --------- CDNA5 async global->LDS staging (guarded, device pass only) -----
#if defined(__gfx1250__) && __has_builtin(__builtin_amdgcn_global_load_async_to_lds_b128)
#define USE_ASYNC_STAGE 1
#else
#define USE_ASYNC_STAGE 0
#endif

#if USE_ASYNC_STAGE
// builtin signature (from hipcc diagnostic): (v4i global*, v4i shared*, Imm, Imm)
#define ASYNC_CP16(ldst, gsrc)                                                   \
  __builtin_amdgcn_global_load_async_to_lds_b128(                                \
      (__attribute__((address_space(1))) v4i*)(                                  \
          (__attribute__((address_space(1))) void*)(gsrc)),                      \
      (__attribute__((address_space(3))) v4i*)(                                  \
          (__attribute__((address_space(3))) void*)(ldst)),                      \
      0, 0)
#if __has_builtin(__builtin_amdgcn_s_wait_asynccnt)
#define WAIT_ASYNC() __builtin_amdgcn_s_wait_asynccnt(0)
#else
#define WAIT_ASYNC() asm volatile("s_wait_asynccnt 0" ::: "memory")
#endif
#endif

// ---------------------------------------------------------------------------
// Kernel 0: transpose W_dec [1024,16384] -> Wt [16384,1024] in workspace.
// ---------------------------------------------------------------------------
__global__ __launch_bounds__(256) void wdec_transpose(const float* __restrict__ W,
                                                      float* __restrict__ Wt) {
  __shared__ float tile[32][33];
  const int f0 = blockIdx.x * 32;
  const int d0 = blockIdx.y * 32;
  const int tx = threadIdx.x;  // 0..31
  const int ty = threadIdx.y;  // 0..7
#pragma unroll
  for (int i = 0; i < 32; i += 8)
    tile[ty + i][tx] = W[(size_t)(d0 + ty + i) * D_DICT + (f0 + tx)];
  __syncthreads();
#pragma unroll
  for (int i = 0; i < 32; i += 8)
    Wt[(size_t)(f0 + ty + i) * D_MODEL + (d0 + tx)] = tile[tx][ty + i];
}

// ---------------------------------------------------------------------------
// Kernel 1: encoder GEMM with V_WMMA_F32_16X16X4_F32.
//   C[M=8192, N=16384] = A[M,K=1024] * B[K,N] + b_enc,  B[k][n] = W_enc[n][k]
// Tile: 128(M) x 256(N) per 256-thread workgroup; 8 waves in a 2(M) x 4(N)
// grid, each wave owns a 64x64 sub-tile = 4x4 frags of 16x16 (128 C VGPRs).
// LDS tiles row-major [row][k] with a 20-float row stride:
//   - each A/B fragment K-pair = ONE aligned ds_load_b64 (no v_mov shuffles)
//   - stride 20 makes the b64 reads bank-conflict-free for both lane halves
//   - contiguous 16B rows match GLOBAL_LOAD_ASYNC_TO_LDS_B128 writes
// Double-buffered: async-load next K-slab while WMMAs consume the current one.
// ---------------------------------------------------------------------------
#define BM 128
#define BN 256
#define BK 16
#define LROW 20  // padded row stride in floats (16 data + 4 pad)

__device__ __forceinline__ void wmma_slab(const float* __restrict__ As,
                                          const float* __restrict__ Bs,
                                          v8f (&c)[4][4], int lo, int hi, int mw,
                                          int nw) {
#pragma unroll
  for (int kk = 0; kk < BK; kk += 4) {
    v2f a[4], b[4];
#pragma unroll
    for (int mi = 0; mi < 4; ++mi)
      a[mi] = *(const v2f*)(As + (mw + mi * 16 + lo) * LROW + kk + hi * 2);
#pragma unroll
    for (int ni = 0; ni < 4; ++ni)
      b[ni] = *(const v2f*)(Bs + (nw + ni * 16 + lo) * LROW + kk + hi * 2);
#pragma unroll
    for (int mi = 0; mi < 4; ++mi)
#pragma unroll
      for (int ni = 0; ni < 4; ++ni)
        c[mi][ni] = __builtin_amdgcn_wmma_f32_16x16x4_f32(
            false, a[mi], false, b[ni], (short)0, c[mi][ni], false, false);
  }
}

__global__ __launch_bounds__(256) void enc_gemm(const float* __restrict__ x,
                                                const float* __restrict__ We,
                                                const float* __restrict__ be,
                                                float* __restrict__ z) {
  __shared__ float Abuf[2][BM * LROW];  // 2 x 10 KB
  __shared__ float Bbuf[2][BN * LROW];  // 2 x 20 KB

  const int t    = threadIdx.x;
  const int lane = t & 31;
  const int wid  = t >> 5;
  const int lo   = lane & 15;   // M (A) / N (B) position within frag
  const int hi   = lane >> 4;   // selects K offset {0,2} within frag
  const int wm   = wid & 1;     // wave row (2)
  const int wn   = wid >> 1;    // wave col (4)
  const int m0   = blockIdx.y * BM;
  const int n0   = blockIdx.x * BN;
  const int mw   = wm * 64;
  const int nw   = wn * 64;

  v8f c[4][4];
#pragma unroll
  for (int mi = 0; mi < 4; ++mi)
#pragma unroll
    for (int ni = 0; ni < 4; ++ni)
#pragma unroll
      for (int e = 0; e < 8; ++e) c[mi][ni][e] = 0.0f;

  // Staging assignments (256 threads):
  //   A tile: 128 rows x 16 k -> thread stages 8 floats: row t>>1, k-half (t&1)*8
  //   B tile: 256 rows x 16 k -> thread stages full row t
  const int alr = t >> 1;
  const int akh = (t & 1) * 8;
  const float* gA0 = x + (size_t)(m0 + alr) * D_MODEL + akh;
  const float* gB0 = We + (size_t)(n0 + t) * D_MODEL;

#if USE_ASYNC_STAGE
  // ---- async double-buffered pipeline ----
  {
    float* lA = &Abuf[0][alr * LROW + akh];
    float* lB = &Bbuf[0][t * LROW];
    ASYNC_CP16(lA, gA0);
    ASYNC_CP16(lA + 4, gA0 + 4);
#pragma unroll
    for (int j = 0; j < 4; ++j) ASYNC_CP16(lB + 4 * j, gB0 + 4 * j);
  }
  for (int k0 = 0; k0 < D_MODEL; k0 += BK) {
    const int cur = (k0 >> 4) & 1;
    WAIT_ASYNC();
    __syncthreads();  // current slab fully in LDS for all waves
    if (k0 + BK < D_MODEL) {
      float* lA = &Abuf[cur ^ 1][alr * LROW + akh];
      float* lB = &Bbuf[cur ^ 1][t * LROW];
      const float* gA = gA0 + k0 + BK;
      const float* gB = gB0 + k0 + BK;
      ASYNC_CP16(lA, gA);
      ASYNC_CP16(lA + 4, gA + 4);
#pragma unroll
      for (int j = 0; j < 4; ++j) ASYNC_CP16(lB + 4 * j, gB + 4 * j);
    }
    wmma_slab(Abuf[cur], Bbuf[cur], c, lo, hi, mw, nw);
  }
#else
  // ---- register-staged double-buffered fallback ----
  {
    const float4* ap = reinterpret_cast<const float4*>(gA0);
    const float4* bp = reinterpret_cast<const float4*>(gB0);
    float4 ra0 = ap[0], ra1 = ap[1];
    float4 rb0 = bp[0], rb1 = bp[1], rb2 = bp[2], rb3 = bp[3];
    *reinterpret_cast<float4*>(&Abuf[0][alr * LROW + akh]) = ra0;
    *reinterpret_cast<float4*>(&Abuf[0][alr * LROW + akh + 4]) = ra1;
    float4* lb = reinterpret_cast<float4*>(&Bbuf[0][t * LROW]);
    lb[0] = rb0; lb[1] = rb1; lb[2] = rb2; lb[3] = rb3;
  }
  for (int k0 = 0; k0 < D_MODEL; k0 += BK) {
    const int cur = (k0 >> 4) & 1;
    __syncthreads();  // current slab ready; previous reads of next buf done
    float4 ra0, ra1, rb0, rb1, rb2, rb3;
    const bool more = (k0 + BK < D_MODEL);
    if (more) {
      const float4* ap = reinterpret_cast<const float4*>(gA0 + k0 + BK);
      const float4* bp = reinterpret_cast<const float4*>(gB0 + k0 + BK);
      ra0 = ap[0]; ra1 = ap[1];
      rb0 = bp[0]; rb1 = bp[1]; rb2 = bp[2]; rb3 = bp[3];
    }
    wmma_slab(Abuf[cur], Bbuf[cur], c, lo, hi, mw, nw);
    if (more) {
      *reinterpret_cast<float4*>(&Abuf[cur ^ 1][alr * LROW + akh]) = ra0;
      *reinterpret_cast<float4*>(&Abuf[cur ^ 1][alr * LROW + akh + 4]) = ra1;
      float4* lb = reinterpret_cast<float4*>(&Bbuf[cur ^ 1][t * LROW]);
      lb[0] = rb0; lb[1] = rb1; lb[2] = rb2; lb[3] = rb3;
    }
  }
#endif

  // Epilogue: + b_enc, store raw z. C frag layout: VGPR e -> row e (lanes
  // 0-15) / row e+8 (lanes 16-31), col = lane % 16.
#pragma unroll
  for (int ni = 0; ni < 4; ++ni) {
    const int col = n0 + nw + ni * 16 + lo;
    const float bias = be[col];
#pragma unroll
    for (int mi = 0; mi < 4; ++mi) {
#pragma unroll
      for (int e = 0; e < 8; ++e) {
        const int r = m0 + mw + mi * 16 + hi * 8 + e;
        z[(size_t)r * D_DICT + col] = c[mi][ni][e] + bias;
      }
    }
  }
}

// ---------------------------------------------------------------------------
// Kernel 2: per-row exact top-32 (radix select on |z| bit pattern) + masked
// writeback + sparse decoder.  One 256-thread workgroup per token row; each
// thread holds 64 row elements in registers (col = i*256 + t, coalesced).
// ---------------------------------------------------------------------------
__global__ __launch_bounds__(256) void topk_decode(float* __restrict__ z,
                                                   const float* __restrict__ Wd,
                                                   const float* __restrict__ Wt,
                                                   const float* __restrict__ bd,
                                                   float* __restrict__ recon,
                                                   int use_wt) {
  const int t = threadIdx.x;
  const int row = blockIdx.x;
  float* zr = z + (size_t)row * D_DICT;

  float v[64];
  unsigned u[64];
#pragma unroll
  for (int i = 0; i < 64; ++i) {
    v[i] = zr[i * 256 + t];
    u[i] = __float_as_uint(v[i]) & 0x7fffffffu;  // |z| bits, monotonic as uint
  }

  __shared__ unsigned hist[256];
  __shared__ unsigned s_pref;
  __shared__ int s_kk;
  __shared__ int scanb[256];
  __shared__ int eqcols[128];
  __shared__ int s_E;
  __shared__ int selcol[64];
  __shared__ float selval[64];

  // ---- 4-pass radix select: find T = 32nd-largest |z| bit pattern ----
  unsigned pref = 0;
  int kk = TOPK;
  for (int pass = 0; pass < 4; ++pass) {
    const int sh = 24 - 8 * pass;
    hist[t] = 0;
    __syncthreads();
#pragma unroll
    for (int i = 0; i < 64; ++i) {
      const unsigned ui = u[i];
      const bool keep = (pass == 0) || ((ui >> (sh + 8)) == pref);
      if (keep) atomicAdd(&hist[(ui >> sh) & 0xffu], 1u);
    }
    __syncthreads();
    if (t == 0) {
      unsigned c = 0;
      int b = 255;
      for (; b > 0; --b) {
        const unsigned h = hist[b];
        if (c + h >= (unsigned)kk) break;
        c += h;
      }
      s_pref = (pref << 8) | (unsigned)b;
      s_kk = kk - (int)c;
    }
    __syncthreads();
    pref = s_pref;
    kk = s_kk;
    __syncthreads();
  }
  const unsigned T = pref;  // exact 32nd-largest |z| bits
  const int kkf = kk;       // how many T-valued elements to take (by low idx)

  // ---- rank ties by column index (matches jax.lax.top_k tie order) ----
  int myeq = 0;
#pragma unroll
  for (int i = 0; i < 64; ++i) myeq += (u[i] == T) ? 1 : 0;
  scanb[t] = myeq;
  __syncthreads();
  for (int off = 1; off < 256; off <<= 1) {
    const int add = (t >= off) ? scanb[t - off] : 0;
    __syncthreads();
    scanb[t] += add;
    __syncthreads();
  }
  const int eq_excl = scanb[t] - myeq;
  const int Etot = scanb[255];
  {
    int pos = eq_excl;
#pragma unroll
    for (int i = 0; i < 64; ++i) {
      if (u[i] == T) {
        if (pos < 128) eqcols[pos] = i * 256 + t;
        ++pos;
      }
    }
  }
  __syncthreads();
  if (t == 0) {
    const int E = Etot < 128 ? Etot : 128;
    for (int a = 1; a < E; ++a) {  // tiny insertion sort (E is ~1 in practice)
      const int key = eqcols[a];
      int b = a - 1;
      while (b >= 0 && eqcols[b] > key) { eqcols[b + 1] = eqcols[b]; --b; }
      eqcols[b + 1] = key;
    }
    s_E = E;
  }
  __syncthreads();
  const int E = s_E;
  const int ntake = kkf < E ? kkf : E;

  // ---- selection predicate + masked writeback + deterministic compaction ----
  unsigned long long selmask = 0ull;
  int mysel = 0;
#pragma unroll
  for (int i = 0; i < 64; ++i) {
    bool s = (u[i] > T);
    if (!s && u[i] == T) {
      const int col = i * 256 + t;
      for (int j = 0; j < ntake; ++j)
        if (eqcols[j] == col) { s = true; break; }
    }
    if (s) {
      selmask |= (1ull << i);
      ++mysel;
    }
  }
  scanb[t] = mysel;
  __syncthreads();
  for (int off = 1; off < 256; off <<= 1) {
    const int add = (t >= off) ? scanb[t - off] : 0;
    __syncthreads();
    scanb[t] += add;
    __syncthreads();
  }
  const int sel_excl = scanb[t] - mysel;
  const int nsel_tot = scanb[255];
  {
    int pos = sel_excl;
#pragma unroll
    for (int i = 0; i < 64; ++i) {
      const bool s = (selmask >> i) & 1ull;
      zr[i * 256 + t] = s ? v[i] : 0.0f;  // masked z output (coalesced)
      if (s) {
        if (pos < 64) { selcol[pos] = i * 256 + t; selval[pos] = v[i]; }
        ++pos;
      }
    }
  }
  __syncthreads();
  const int nsel = nsel_tot < 64 ? nsel_tot : 64;  // == 32 by construction

  // ---- sparse decoder: recon = b_dec + sum_j z_j * W_dec[:, idx_j] ----
  float acc[4];
#pragma unroll
  for (int r = 0; r < 4; ++r) acc[r] = bd[t + r * 256];
  for (int j = 0; j < nsel; ++j) {
    const int cj = selcol[j];
    const float vj = selval[j];
    if (use_wt) {
      const float* wp = Wt + (size_t)cj * D_MODEL;  // contiguous 4KB (L2-hot)
#pragma unroll
      for (int r = 0; r < 4; ++r) acc[r] += vj * wp[t + r * 256];
    } else {
#pragma unroll
      for (int r = 0; r < 4; ++r)
        acc[r] += vj * Wd[(size_t)(t + r * 256) * D_DICT + cj];
    }
  }
  float* rr = recon + (size_t)row * D_MODEL;
#pragma unroll
  for (int r = 0; r < 4; ++r) rr[t + r * 256] = acc[r];
}

// ---------------------------------------------------------------------------
extern "C" void kernel_launch(void* const* d_in, const int* in_sizes, int n_in,
                              void* d_out, int out_size, void* d_ws, size_t ws_size,
                              hipStream_t stream) {
  const float* x  = (const float*)d_in[0];
  const float* We = (const float*)d_in[1];
  const float* be = (const float*)d_in[2];
  const float* Wd = (const float*)d_in[3];
  const float* bd = (const float*)d_in[4];

  float* recon = (float*)d_out;
  float* z     = recon + (size_t)N_TOKENS * D_MODEL;

  float* wt = (float*)d_ws;
  const int use_wt = (ws_size >= (size_t)D_DICT * D_MODEL * sizeof(float)) ? 1 : 0;

  if (use_wt) {
    wdec_transpose<<<dim3(D_DICT / 32, D_MODEL / 32), dim3(32, 8), 0, stream>>>(Wd, wt);
  }
  enc_gemm<<<dim3(D_DICT / BN, N_TOKENS / BM), 256, 0, stream>>>(x, We, be, z);
  topk_decode<<<dim3(N_TOKENS), 256, 0, stream>>>(z, Wd, wt, bd, recon, use_wt);
}